// Encoder_20822001451484
// MI455X (gfx1250) — compile-verified
//
#include <hip/hip_runtime.h>
#include <hip/hip_bf16.h>

// ---------------------------------------------------------------------------
// Problem constants (from the reference)
// ---------------------------------------------------------------------------
#define NUM_AUTHOR 16604
#define D          300            // embed dim
#define DH         150            // D/2 (float2 lanes)
#define K3         900            // 3*D, GEMM K
#define W1T_LD     304            // padded N dim for transposed W1 (19*16)
#define APANEL     (K3 * 16)      // floats per 16-row A panel = 14400
#define BROW       (W1T_LD * 2)   // floats per K-pair row of W1T_sw = 608

typedef float v2f __attribute__((ext_vector_type(2)));
typedef float v8f __attribute__((ext_vector_type(8)));

// ---------------------------------------------------------------------------
// Kernel 0: W1 [300 x 900] row-major -> W1T_sw in K-pair fragment order:
//   W1T_sw[(k>>1)*608 + j*2 + (k&1)] = W1[j*900 + k]   (j >= 300 -> 0)
// A lane's two K values are adjacent -> B fragment = one coalesced b64 load.
// ---------------------------------------------------------------------------
__global__ __launch_bounds__(256) void transpose_w1_kernel(
    const float* __restrict__ W1, float* __restrict__ W1T_sw) {
    int idx = blockIdx.x * blockDim.x + threadIdx.x;
    if (idx >= (K3 / 2) * BROW) return;
    int kp  = idx / BROW;
    int rem = idx - kp * BROW;
    int j   = rem >> 1;
    int r   = rem & 1;
    int k   = (kp << 1) + r;
    W1T_sw[idx] = (j < D) ? W1[j * K3 + k] : 0.0f;
}

// ---------------------------------------------------------------------------
// lower_bound over the sorted segment array
// ---------------------------------------------------------------------------
__device__ __forceinline__ int lower_bound_i32(const int* __restrict__ a, int n, int v) {
    int lo = 0, hi = n;
    while (lo < hi) {
        int mid = (lo + hi) >> 1;
        if (a[mid] < v) lo = mid + 1; else hi = mid;
    }
    return lo;
}

// ---------------------------------------------------------------------------
// Kernel 1: gather + segment mean-pool + build WMMA fragment-order A panel.
//
// A_sw layout (16-row panels, K-pair order):
//   A_sw[(b>>4)*14400 + (k>>1)*32 + (b&15)*2 + (k&1)]
//   k   0..299 : self embedding
//   k 300..599 : t1 (same-type mean, 1.0 fallback)
//   k 600..899 : t2 (other-type mean, 1.0 fallback)
//
// 150 compute threads x float2: halves gather instruction count; dominant
// traffic (E*D*4 = 1.26 GB) hits the L2-resident 35 MB embedding table.
// ---------------------------------------------------------------------------
#define POOL_THREADS 256
#define CHUNK 256

__global__ __launch_bounds__(POOL_THREADS) void pool_kernel(
    const int*   __restrict__ nodes,
    const int*   __restrict__ seg,
    const int*   __restrict__ neighbors,
    const float* __restrict__ emb,
    float*       __restrict__ A_sw,
    int E) {
    __shared__ int   s_nbr[CHUNK];
    __shared__ float s_same[CHUNK];

    const int b = blockIdx.x;
    const int t = threadIdx.x;

    const int  node       = nodes[b];
    const bool selfAuthor = (node < NUM_AUTHOR);

    const int start = lower_bound_i32(seg, E, b);
    const int end   = lower_bound_i32(seg, E, b + 1);

    v2f  sum1 = {0.0f, 0.0f}, sum2 = {0.0f, 0.0f};
    float c1 = 0.0f, c2 = 0.0f;

    for (int base = start; base < end; base += CHUNK) {
        const int n = min(CHUNK, end - base);
        __syncthreads();
        if (t < n) {
            int nb = neighbors[base + t];
            s_nbr[t]  = nb;
            s_same[t] = ((nb < NUM_AUTHOR) == selfAuthor) ? 1.0f : 0.0f;
        }
        __syncthreads();
        if (t < DH) {
            for (int i = 0; i < n; ++i) {
                const float w = s_same[i];
                const float u = 1.0f - w;
                const v2f  v = *(const v2f*)(emb + s_nbr[i] * D + 2 * t);  // coalesced b64
                sum1 += w * v;
                sum2 += u * v;
                c1   += w;
                c2   += u;
            }
        }
    }

    if (t < DH) {
        v2f t1, t2;
        t1.x = (c1 > 0.0f) ? (sum1.x / c1) : 1.0f;
        t1.y = (c1 > 0.0f) ? (sum1.y / c1) : 1.0f;
        t2.x = (c2 > 0.0f) ? (sum2.x / c2) : 1.0f;
        t2.y = (c2 > 0.0f) ? (sum2.y / c2) : 1.0f;
        const v2f sv = *(const v2f*)(emb + node * D + 2 * t);

        // dims (2t, 2t+1) == K-pair t (self), 150+t (t1), 300+t (t2)
        float* dst = A_sw + (size_t)(b >> 4) * APANEL + (b & 15) * 2;
        *(v2f*)(dst + (t      ) * 32) = sv;
        *(v2f*)(dst + (t + 150) * 32) = t1;
        *(v2f*)(dst + (t + 300) * 32) = t2;
    }
}

// ---------------------------------------------------------------------------
// Kernel 2: out[b,j] = sum_k A[b,k]*W1[j,k] + b1[j]  via V_WMMA_F32_16X16X4_F32
//
// One wave -> one 16x16 tile; 225 K-steps of 4. Fragment-order staging makes
// each A/B fragment a single fully-coalesced global_load_b64 (256B/fragment).
// A frag (ISA 7.12.2): lanes 0-15 K={k0,k0+1}, lanes 16-31 K={k0+2,k0+3}
//   -> lane's K-pair index kp = k0/2 + (lane>=16).
// D tile: VGPR i -> row (i + 8*(lane>=16)), col (lane&15). Bias in C init.
// ---------------------------------------------------------------------------
__global__ __launch_bounds__(256) void gemm_wmma_kernel(
    const float* __restrict__ A_sw,
    const float* __restrict__ W1T_sw,
    const float* __restrict__ b1,
    float*       __restrict__ out) {
    const int wave = threadIdx.x >> 5;
    const int lane = threadIdx.x & 31;
    const int nl   = lane & 15;          // M index (A frag) / N index (B frag)
    const int hi   = lane >> 4;          // lane-half select

    const int n0    = blockIdx.x * 16;             // N tile (19 tiles, last padded)
    const int mtile = blockIdx.y * 8 + wave;       // 16-row M tile index
    const int col   = n0 + nl;

    const float* Ap = A_sw   + (size_t)mtile * APANEL + nl * 2;
    const float* Bp = W1T_sw + col * 2;

    const float bias = (col < D) ? b1[col] : 0.0f;
    v8f c = {bias, bias, bias, bias, bias, bias, bias, bias};

#pragma unroll 5
    for (int k0 = 0; k0 < K3; k0 += 4) {
        const int kp = (k0 >> 1) + hi;             // this lane's K-pair
        const v2f a  = *(const v2f*)(Ap + kp * 32);    // one b64, 256B/frag
        const v2f bb = *(const v2f*)(Bp + kp * BROW);  // one b64, cache-hot
        c = __builtin_amdgcn_wmma_f32_16x16x4_f32(
                /*neg_a=*/false, a, /*neg_b=*/false, bb,
                /*c_mod=*/(short)0, c, /*reuse_a=*/false, /*reuse_b=*/false);
    }

    if (col < D) {
        const int row0 = mtile * 16 + (hi << 3);
#pragma unroll
        for (int i = 0; i < 8; ++i) {
            out[(row0 + i) * D + col] = c[i];
        }
    }
}

// ---------------------------------------------------------------------------
// Launch
// ---------------------------------------------------------------------------
extern "C" void kernel_launch(void* const* d_in, const int* in_sizes, int n_in,
                              void* d_out, int out_size, void* d_ws, size_t ws_size,
                              hipStream_t stream) {
    const int*   nodes     = (const int*)d_in[0];
    const int*   seg       = (const int*)d_in[1];
    const int*   neighbors = (const int*)d_in[2];
    const float* emb       = (const float*)d_in[3];
    const float* W1        = (const float*)d_in[4];
    const float* b1        = (const float*)d_in[5];
    float*       out       = (float*)d_out;

    const int Bn = in_sizes[0];   // 32768
    const int E  = in_sizes[1];   // 1048576

    // Workspace: A panels [B/16][450][16][2] floats, then W1T_sw [450][304][2].
    float* A_sw   = (float*)d_ws;
    float* W1T_sw = A_sw + (size_t)Bn * K3;

    // 0) transpose/pad W1 into K-pair fragment order
    {
        int total  = (K3 / 2) * BROW;
        int blocks = (total + 255) / 256;
        transpose_w1_kernel<<<blocks, 256, 0, stream>>>(W1, W1T_sw);
    }
    // 1) gather + segment mean-pool -> A panels
    pool_kernel<<<Bn, POOL_THREADS, 0, stream>>>(nodes, seg, neighbors, emb, A_sw, E);

    // 2) WMMA f32 GEMM: grid = (19 N-tiles, B/128 M-blocks), 8 waves/block
    dim3 grid((D + 15) / 16, Bn / 128);
    gemm_wmma_kernel<<<grid, 256, 0, stream>>>(A_sw, W1T_sw, b1, out);
}